// Router_64381559767962
// MI455X (gfx1250) — compile-verified
//
#include <hip/hip_runtime.h>

typedef float v2f __attribute__((ext_vector_type(2)));
typedef float v8f __attribute__((ext_vector_type(8)));

constexpr int DIM  = 4096;   // reduction dim
constexpr int NE   = 16;     // experts
constexpr int KCH  = 512;    // K-chunk staged in LDS
constexpr int WROW = KCH + 4;    // padded LDS row stride (floats) -> bank-conflict free
constexpr int TPB  = 128;    // tokens per block (8 waves x 16 tokens)

__global__ __launch_bounds__(256) void router_kernel(
    const float* __restrict__ x,
    const float* __restrict__ w1,
    const float* __restrict__ b1,
    const float* __restrict__ rb,
    float* __restrict__ outV,
    int*   __restrict__ outI)
{
    __shared__ float sW[NE * WROW];      // 33,024 B
    __shared__ float sLog[TPB * 17];     //  8,704 B

    const int tid  = threadIdx.x;
    const int lane = tid & 31;
    const int wave = tid >> 5;
    const int row  = lane & 15;          // token-in-tile (A) / expert (B)
    const int koff = (lane >> 4) << 1;   // 0 or 2 (K sub-offset per lane half)
    const int tokenBase = blockIdx.x * TPB + wave * 16;

    const float* xp = x + (size_t)(tokenBase + row) * DIM + koff;

    v8f c = {};
    for (int chunk = 0; chunk < DIM / KCH; ++chunk) {
        const int k0 = chunk * KCH;
        __syncthreads();                 // protect sW from prior readers
        // cooperative stage of W[:, k0:k0+KCH] into LDS (coalesced float4)
        #pragma unroll
        for (int r = 0; r < 8; ++r) {
            const int i = tid + (r << 8);        // 0..2047 float4 slots
            const int e = i >> 7;                // expert row
            const int j = (i & 127) << 2;        // float offset within chunk
            *(float4*)&sW[e * WROW + j] = *(const float4*)(w1 + e * DIM + k0 + j);
        }
        __syncthreads();

        const float* xc = xp + k0;
        #pragma unroll 8
        for (int kk = 0; kk < KCH; kk += 4) {
            v2f a = __builtin_nontemporal_load((const v2f*)(xc + kk));
            v2f b = *(const v2f*)&sW[row * WROW + koff + kk];
            c = __builtin_amdgcn_wmma_f32_16x16x4_f32(
                    /*neg_a=*/false, a, /*neg_b=*/false, b,
                    /*c_mod=*/(short)0, c, /*reuse_a=*/false, /*reuse_b=*/false);
        }
    }

    // Scatter logits: lane holds expert (lane&15) for tokens v + 8*(lane>>4)
    {
        const int e    = lane & 15;
        const int toff = wave * 16 + ((lane >> 4) << 3);
        #pragma unroll
        for (int v = 0; v < 8; ++v)
            sLog[(toff + v) * 17 + e] = c[v];
    }
    __syncthreads();

    if (tid < TPB) {
        const int t    = tid;
        const int gtok = blockIdx.x * TPB + t;

        float sc[NE];
        float mx = -3.402823466e38f;
        #pragma unroll
        for (int e = 0; e < NE; ++e) {
            float l = sLog[t * 17 + e] + b1[e];   // logits + w1_bias
            sc[e] = l;
            mx = fmaxf(mx, l);
        }
        float sum = 0.f;
        #pragma unroll
        for (int e = 0; e < NE; ++e) { float ex = __expf(sc[e] - mx); sc[e] = ex; sum += ex; }
        const float inv = 1.0f / sum;
        #pragma unroll
        for (int e = 0; e < NE; ++e) sc[e] = sc[e] * inv + rb[e];   // scores + route_bias

        // group max over 4 groups of 4
        float gs[4];
        #pragma unroll
        for (int g = 0; g < 4; ++g)
            gs[g] = fmaxf(fmaxf(sc[4*g], sc[4*g+1]), fmaxf(sc[4*g+2], sc[4*g+3]));

        // top-2 groups, first-occurrence argmax (matches top_k tie-break)
        int g1 = 0; float gv1 = gs[0];
        #pragma unroll
        for (int g = 1; g < 4; ++g) if (gs[g] > gv1) { gv1 = gs[g]; g1 = g; }
        int g2 = -1; float gv2 = -3.402823466e38f;
        #pragma unroll
        for (int g = 0; g < 4; ++g) if (g != g1 && gs[g] > gv2) { gv2 = gs[g]; g2 = g; }

        // top-2 experts over group-masked scores
        const float NEG = -3.402823466e38f;
        int i1 = 0; float v1 = NEG;
        #pragma unroll
        for (int e = 0; e < NE; ++e) {
            const int gi = e >> 2;
            const float val = (gi == g1 || gi == g2) ? sc[e] : NEG;
            if (val > v1) { v1 = val; i1 = e; }
        }
        int i2 = 0; float v2 = NEG;
        #pragma unroll
        for (int e = 0; e < NE; ++e) {
            const int gi = e >> 2;
            const float val = ((gi == g1 || gi == g2) && e != i1) ? sc[e] : NEG;
            if (val > v2) { v2 = val; i2 = e; }
        }

        // values gathered from original (unmasked) scores; ROUTE_SCALE = 1.0
        outV[(size_t)gtok * 2 + 0] = sc[i1];
        outV[(size_t)gtok * 2 + 1] = sc[i2];
        outI[(size_t)gtok * 2 + 0] = i1;
        outI[(size_t)gtok * 2 + 1] = i2;
    }
}

extern "C" void kernel_launch(void* const* d_in, const int* in_sizes, int n_in,
                              void* d_out, int out_size, void* d_ws, size_t ws_size,
                              hipStream_t stream) {
    (void)n_in; (void)out_size; (void)d_ws; (void)ws_size;
    const float* x  = (const float*)d_in[0];
    const float* w1 = (const float*)d_in[1];
    const float* b1 = (const float*)d_in[2];
    const float* rb = (const float*)d_in[3];
    const int n = in_sizes[0] / DIM;          // 65536 tokens

    float* outV = (float*)d_out;                       // [n,2] fp32 values
    int*   outI = (int*)(outV + (size_t)n * 2);        // [n,2] int32 indices

    dim3 grid(n / TPB), block(256);
    router_kernel<<<grid, block, 0, stream>>>(x, w1, b1, rb, outV, outI);
}